// MPSClassifier_5652176962111
// MI455X (gfx1250) — compile-verified
//
#include <hip/hip_runtime.h>

typedef float v8f __attribute__((ext_vector_type(8)));
typedef float v2f __attribute__((ext_vector_type(2)));

#define BATCH  512
#define MSIZE  256
#define DDIM   32
#define NLAB   10
#define NSEG   4
#define SEGLEN (MSIZE / NSEG)

__device__ __forceinline__ v8f wmma4(v2f a, v2f b, v8f c) {
  // V_WMMA_F32_16X16X4_F32 : D = A(16x4) x B(4x16) + C(16x16), fp32
  return __builtin_amdgcn_wmma_f32_16x16x4_f32(
      /*neg_a=*/false, a, /*neg_b=*/false, b,
      /*c_mod=*/(short)0, c, /*reuse_a=*/false, /*reuse_b=*/false);
}

// Sout[ti][tj] = sum_tk A[ti][tk] x Sin[tk][tj].
// S tiles in C-layout (reg=row-in-tile e+8h, lane=col-in-tile m). B operand for
// K-chunk v is elements {2v,2v+1}: hardware k = r + 2h -> tile rows {2v+r+8h};
// A operands are built with the identical (permuted) K ordering.
__device__ __forceinline__ void mm_step(v8f (&Sout)[2][2], const v8f (&Sin)[2][2],
                                        const v2f (&A)[2][2][4]) {
  #pragma unroll
  for (int ti = 0; ti < 2; ++ti)
    #pragma unroll
    for (int tj = 0; tj < 2; ++tj) {
      v8f acc = {0.f,0.f,0.f,0.f,0.f,0.f,0.f,0.f};
      #pragma unroll
      for (int tk = 0; tk < 2; ++tk)
        #pragma unroll
        for (int v = 0; v < 4; ++v) {
          v2f bop;
          bop.x = Sin[tk][tj][2 * v];
          bop.y = Sin[tk][tj][2 * v + 1];
          acc = wmma4(A[ti][tk][v], bop, acc);
        }
      Sout[ti][tj] = acc;
    }
}

// Build A = M_s^T in A layout. Lane holds column (16ti+m) of M; reg r of
// K-chunk v holds M[16tk + 2v + r + 8h, 16ti+m]. One b64 load fetches the
// (d=0,d=1) core pair, blended with packed FMAs.
__device__ __forceinline__ void load_leaf(v2f (&A)[2][2][4],
                                          const float* __restrict__ cs,
                                          float w0, float w1, int m, int h) {
  #pragma unroll
  for (int ti = 0; ti < 2; ++ti) {
    const int c = 16 * ti + m;
    #pragma unroll
    for (int tk = 0; tk < 2; ++tk)
      #pragma unroll
      for (int v = 0; v < 4; ++v) {
        const int r0 = 16 * tk + 2 * v + 8 * h;
        const v2f f0 = *(const v2f*)(cs + (r0 * DDIM + c) * 2);
        const v2f f1 = *(const v2f*)(cs + ((r0 + 1) * DDIM + c) * 2);
        v2f a;
        a.x = w0 * f0.x + w1 * f0.y;
        a.y = w0 * f1.x + w1 * f1.y;
        A[ti][tk][v] = a;
      }
  }
}

__global__ __launch_bounds__(NSEG * 32, 1)
void mps_chain_kernel(const float* __restrict__ x,      // [512,256]
                      const float* __restrict__ core,   // [256,32,32,2]
                      const float* __restrict__ label,  // [32,32,10]
                      float* __restrict__ out)          // [512,10]
{
  __shared__ float pmat[NSEG][DDIM][DDIM];   // 16 KB: segment partial products

  const int b    = blockIdx.x;
  const int wave = (int)(threadIdx.x >> 5);  // segment id
  const int lane = (int)(threadIdx.x & 31);
  const int m    = lane & 15;                // col-within-tile / A-row
  const int h    = lane >> 4;                // lane half

  // Sa = I  (C-layout tiles: element e -> row 16tk+e+8h, col 16tj+m)
  v8f Sa[2][2], Sb[2][2];
  #pragma unroll
  for (int tk = 0; tk < 2; ++tk)
    #pragma unroll
    for (int tj = 0; tj < 2; ++tj)
      #pragma unroll
      for (int e = 0; e < 8; ++e) {
        const int row = 16 * tk + e + 8 * h;
        const int col = 16 * tj + m;
        Sa[tk][tj][e] = (row == col) ? 1.0f : 0.0f;
      }

  const float* xb = x + b * MSIZE;
  const int pbase = wave * SEGLEN;

  // leaf index for chain position t within this segment (bit-reversed order)
  #define LEAF(t) ((int)(__brev((unsigned)(pbase + (t))) >> 24))

  // Software-pipelined chain: S <- M_sigma(p)^T @ S, ping-pong Sa/Sb (no copies)
  v2f A0[2][2][4], A1[2][2][4];
  {
    const int s0 = LEAF(0);
    const float w1 = xb[s0];
    load_leaf(A0, core + s0 * (DDIM * DDIM * 2), 1.0f - w1, w1, m, h);
  }

  for (int t = 0; t < SEGLEN; t += 2) {
    {
      const int s1 = LEAF(t + 1);
      const float w1 = xb[s1];
      load_leaf(A1, core + s1 * (DDIM * DDIM * 2), 1.0f - w1, w1, m, h);
    }
    // warm WGP$ two leaves ahead (emits global_prefetch_b8)
    __builtin_prefetch(core + LEAF((t + 4) & (SEGLEN - 1)) * (DDIM * DDIM * 2), 0, 0);

    mm_step(Sb, Sa, A0);                       // 32x v_wmma_f32_16x16x4_f32

    {
      const int s2 = LEAF((t + 2) & (SEGLEN - 1));   // wraps on last iter (harmless)
      const float w1 = xb[s2];
      load_leaf(A0, core + s2 * (DDIM * DDIM * 2), 1.0f - w1, w1, m, h);
    }

    mm_step(Sa, Sb, A1);
  }
  #undef LEAF
  // segment partial P_w now in Sa

  // publish partial P_w (row-major) to LDS
  #pragma unroll
  for (int tk = 0; tk < 2; ++tk)
    #pragma unroll
    for (int tj = 0; tj < 2; ++tj)
      #pragma unroll
      for (int e = 0; e < 8; ++e)
        pmat[wave][16 * tk + e + 8 * h][16 * tj + m] = Sa[tk][tj][e];

  __syncthreads();

  if (wave == 0) {
    // combine: S <- P_w @ S for w = 1..3  (S_total = P3 P2 P1 P0), ping-pong
    v2f A[2][2][4];
    #pragma unroll
    for (int w = 1; w < NSEG; ++w) {
      #pragma unroll
      for (int ti = 0; ti < 2; ++ti)
        #pragma unroll
        for (int tk = 0; tk < 2; ++tk)
          #pragma unroll
          for (int v = 0; v < 4; ++v) {
            const int k0 = 16 * tk + 2 * v + 8 * h;
            // contiguous, 8B aligned -> ds_load_b64
            A[ti][tk][v] = *(const v2f*)&pmat[w][16 * ti + m][k0];  // A = P_w (no ^T)
          }
      if (w & 1) mm_step(Sb, Sa, A);
      else       mm_step(Sa, Sb, A);
    }
    // w=1 -> Sb, w=2 -> Sa, w=3 -> Sb : final product lives in Sb

    // scores[b,l] = sum_{j,i} S[j,i] * L[j,i,l]   (S = R^T -> elementwise dot)
    float acc[NLAB];
    #pragma unroll
    for (int l = 0; l < NLAB; ++l) acc[l] = 0.0f;

    #pragma unroll
    for (int tk = 0; tk < 2; ++tk)
      #pragma unroll
      for (int tj = 0; tj < 2; ++tj)
        #pragma unroll
        for (int e = 0; e < 8; ++e) {
          const int j = 16 * tk + e + 8 * h;
          const int i = 16 * tj + m;
          const float sv = Sb[tk][tj][e];
          const float* lp = label + (j * DDIM + i) * NLAB;
          #pragma unroll
          for (int l = 0; l < NLAB; ++l)
            acc[l] += sv * lp[l];
        }

    #pragma unroll
    for (int l = 0; l < NLAB; ++l) {
      float vsum = acc[l];
      #pragma unroll
      for (int off = 16; off > 0; off >>= 1)
        vsum += __shfl_xor(vsum, off, 32);      // wave32 butterfly
      if (lane == 0) out[b * NLAB + l] = vsum;
    }
  }
}

extern "C" void kernel_launch(void* const* d_in, const int* in_sizes, int n_in,
                              void* d_out, int out_size, void* d_ws, size_t ws_size,
                              hipStream_t stream) {
  (void)in_sizes; (void)n_in; (void)d_ws; (void)ws_size; (void)out_size;
  const float* x     = (const float*)d_in[0];
  const float* core  = (const float*)d_in[1];
  const float* label = (const float*)d_in[2];
  float* out = (float*)d_out;

  dim3 grid(BATCH);
  dim3 block(NSEG * 32);
  hipLaunchKernelGGL(mps_chain_kernel, grid, block, 0, stream, x, core, label, out);
}